// LocalAggBlock_7902739825271
// MI455X (gfx1250) — compile-verified
//
#include <hip/hip_runtime.h>

typedef __attribute__((ext_vector_type(2))) float v2f;
typedef __attribute__((ext_vector_type(4))) float v4f;
typedef __attribute__((ext_vector_type(8))) float v8f;

#define NPTS 8192
#define CCH  64
#define KNN  16
#define KSTR 132  // LDS k-stride (floats) for transposed W: 132 % 64 == 4 -> conflict-free halves
#define XSTR 68   // LDS row stride (floats) for per-point diff rows (64 feat-diff + 4 coord-diff)
#define WAVE_LDS (2*64 + 2*16*XSTR)   // fibuf0, fibuf1, X0, X1 per wave (floats)

// ---------------------------------------------------------------------------
// Kernel 1: exact KNN (K=16) per batch. coords+norms staged in LDS (128KB).
// One thread per query; sorted 16-entry insertion list in registers.
// ---------------------------------------------------------------------------
__global__ __launch_bounds__(256) void knn_kernel(const float* __restrict__ coords,
                                                  int* __restrict__ idx_out) {
    __shared__ v4f pts[NPTS];                       // 128 KB
    const int b     = blockIdx.x >> 5;              // 32 blocks per batch
    const int qbase = (blockIdx.x & 31) << 8;
    const float* cb = coords + b * NPTS * 3;

    for (int j = threadIdx.x; j < NPTS; j += 256) {
        float x = cb[j * 3 + 0], y = cb[j * 3 + 1], z = cb[j * 3 + 2];
        v4f p; p.x = x; p.y = y; p.z = z; p.w = x * x + y * y + z * z;
        pts[j] = p;
    }
    __syncthreads();

    const int q = qbase + threadIdx.x;
    const v4f me = pts[q];

    float best[KNN]; int bidx[KNN];
#pragma unroll
    for (int t = 0; t < KNN; ++t) { best[t] = 3.4e38f; bidx[t] = 0; }

    for (int j = 0; j < NPTS; ++j) {
        v4f p = pts[j];                              // uniform addr -> LDS broadcast
        float dot = me.x * p.x;
        dot = fmaf(me.y, p.y, dot);
        dot = fmaf(me.z, p.z, dot);
        float d = fmaf(-2.0f, dot, me.w + p.w);      // same formula as reference
        d = fmaxf(d, 0.0f);
        if (d < best[KNN - 1]) {                     // rare path
            float cd = d; int ci = j;
#pragma unroll
            for (int t = 0; t < KNN; ++t) {
                if (cd < best[t]) {
                    float tf = best[t]; best[t] = cd; cd = tf;
                    int   ti = bidx[t]; bidx[t] = ci; ci = ti;
                }
            }
        }
    }

    int* op = idx_out + (b * NPTS + q) * KNN;
#pragma unroll
    for (int t = 0; t < KNN; ++t) op[t] = bidx[t];
}

// ---------------------------------------------------------------------------
// Kernel 2: per point D(16x64) = X(16x131) @ W(131x64); relu; col-max over 16.
// x = [fi | nb_f - fi | nb_p - pi]. fi block is row-constant -> fibuf.
// W stored TRANSPOSED in LDS so each B-fragment is one aligned ds_load_b64
// (lands in a contiguous even VGPR pair -> no v_mov shuffling before WMMA).
// V_WMMA_F32_16X16X4_F32, 33 k-steps x 4 n-tiles, two points per wave pass.
// ---------------------------------------------------------------------------
__global__ __launch_bounds__(256) void agg_kernel(const float* __restrict__ coords,
                                                  const float* __restrict__ feat,
                                                  const float* __restrict__ Wm,
                                                  const float* __restrict__ bias,
                                                  const int*   __restrict__ idx,
                                                  float* __restrict__ out) {
    __shared__ float LWT[64 * KSTR];        // W transposed: LWT[n*KSTR + k], k=0..130, k=131 zero
    __shared__ float XS[8 * WAVE_LDS];      // per-wave staging

    for (int i = threadIdx.x; i < 131 * 64; i += 256) {
        int k = i >> 6, n = i & 63;
        LWT[n * KSTR + k] = Wm[i];
    }
    if (threadIdx.x < 64) LWT[threadIdx.x * KSTR + 131] = 0.0f;
    __syncthreads();

    const int lane = threadIdx.x & 31;
    const int wav  = threadIdx.x >> 5;
    const int nl   = lane & 15;             // N column / A row within half
    const int h    = lane >> 4;             // lane half -> K offset 2*h

    float* fib0 = &XS[wav * WAVE_LDS];
    float* fib1 = fib0 + 64;
    float* X0   = fib1 + 64;
    float* X1   = X0 + 16 * XSTR;

    // per-lane base row of transposed W for each n-tile
    const float* wt0 = LWT + (nl     ) * KSTR + 2 * h;
    const float* wt1 = LWT + (nl + 16) * KSTR + 2 * h;
    const float* wt2 = LWT + (nl + 32) * KSTR + 2 * h;
    const float* wt3 = LWT + (nl + 48) * KSTR + 2 * h;

    float bs[4];
#pragma unroll
    for (int nt = 0; nt < 4; ++nt) bs[nt] = bias[nt * 16 + nl];

    const int gw = blockIdx.x * 8 + wav;    // 512 waves, 32 points each

    for (int it = 0; it < 16; ++it) {
        const int P0 = gw * 32 + it * 2;    // pair of points, same batch
        const int b  = P0 >> 13;
        const int n0 = P0 & (NPTS - 1);
        const int n1 = n0 + 1;
        const float* fb = feat   + b * NPTS * CCH;
        const float* cb = coords + b * NPTS * 3;

        // lanes 0-15 own rows of point0, lanes 16-31 own rows of point1
        const int myn = h ? n1 : n0;
        const int nbv = idx[(b * NPTS + myn) * KNN + nl];

        v2f fi0 = *(const v2f*)(fb + n0 * CCH + 2 * lane);
        v2f fi1 = *(const v2f*)(fb + n1 * CCH + 2 * lane);
        *(v2f*)(fib0 + 2 * lane) = fi0;
        *(v2f*)(fib1 + 2 * lane) = fi1;

        // gather 32 neighbor rows (coalesced 256B rows), subtract fi, stage diffs
#pragma unroll
        for (int m = 0; m < 32; ++m) {
            int nb = __shfl(nbv, m, 32);
            v2f g  = *(const v2f*)(fb + nb * CCH + 2 * lane);
            if (m < 16) { v2f d0 = g - fi0; *(v2f*)(X0 + m * XSTR + 2 * lane) = d0; }
            else        { v2f d1 = g - fi1; *(v2f*)(X1 + (m - 16) * XSTR + 2 * lane) = d1; }
        }

        // coord diffs: each lane handles its own row
        {
            const float* pc  = cb + myn * 3;
            const float* ncp = cb + nbv * 3;
            v4f dp; dp.x = ncp[0] - pc[0]; dp.y = ncp[1] - pc[1];
                    dp.z = ncp[2] - pc[2]; dp.w = 0.0f;
            float* xr = (h ? X1 : X0) + nl * XSTR + 64;
            *(v4f*)xr = dp;
        }

        v8f a0c[4], a1c[4];
        const v8f zc = {0.f, 0.f, 0.f, 0.f, 0.f, 0.f, 0.f, 0.f};
#pragma unroll
        for (int nt = 0; nt < 4; ++nt) { a0c[nt] = zc; a1c[nt] = zc; }

#define STEP(K0, A0, A1)                                                          \
        {                                                                         \
            v2f b0 = *(const v2f*)(wt0 + (K0));                                   \
            v2f b1 = *(const v2f*)(wt1 + (K0));                                   \
            v2f b2 = *(const v2f*)(wt2 + (K0));                                   \
            v2f b3 = *(const v2f*)(wt3 + (K0));                                   \
            a0c[0] = __builtin_amdgcn_wmma_f32_16x16x4_f32(                       \
                false, (A0), false, b0, (short)0, a0c[0], false, false);          \
            a1c[0] = __builtin_amdgcn_wmma_f32_16x16x4_f32(                       \
                false, (A1), false, b0, (short)0, a1c[0], false, false);          \
            a0c[1] = __builtin_amdgcn_wmma_f32_16x16x4_f32(                       \
                false, (A0), false, b1, (short)0, a0c[1], false, false);          \
            a1c[1] = __builtin_amdgcn_wmma_f32_16x16x4_f32(                       \
                false, (A1), false, b1, (short)0, a1c[1], false, false);          \
            a0c[2] = __builtin_amdgcn_wmma_f32_16x16x4_f32(                       \
                false, (A0), false, b2, (short)0, a0c[2], false, false);          \
            a1c[2] = __builtin_amdgcn_wmma_f32_16x16x4_f32(                       \
                false, (A1), false, b2, (short)0, a1c[2], false, false);          \
            a0c[3] = __builtin_amdgcn_wmma_f32_16x16x4_f32(                       \
                false, (A0), false, b3, (short)0, a0c[3], false, false);          \
            a1c[3] = __builtin_amdgcn_wmma_f32_16x16x4_f32(                       \
                false, (A1), false, b3, (short)0, a1c[3], false, false);          \
        }

        // k-block 0..63: fi (row-constant A, broadcast from fibuf)
#pragma unroll
        for (int ks = 0; ks < 16; ++ks) {
            int k0 = 4 * ks;                // W k-index base (lane adds 2*h via wt*)
            v2f A0 = *(const v2f*)(fib0 + k0 + 2 * h);
            v2f A1 = *(const v2f*)(fib1 + k0 + 2 * h);
            STEP(k0, A0, A1);
        }
        // k-block 64..127: nb_f - fi
#pragma unroll
        for (int ks = 0; ks < 16; ++ks) {
            int k0 = 64 + 4 * ks;
            int xo = nl * XSTR + 4 * ks + 2 * h;
            v2f A0 = *(const v2f*)(X0 + xo);
            v2f A1 = *(const v2f*)(X1 + xo);
            STEP(k0, A0, A1);
        }
        // k-block 128..131: nb_p - pi (W k=131 is zero)
        {
            int k0 = 128;
            int xo = nl * XSTR + 64 + 2 * h;
            v2f A0 = *(const v2f*)(X0 + xo);
            v2f A1 = *(const v2f*)(X1 + xo);
            STEP(k0, A0, A1);
        }
#undef STEP

        // epilogue: max over 16 rows (8 VGPRs + cross-half), bias + relu, store
        float* outp = out + (size_t)P0 * CCH;
#pragma unroll
        for (int nt = 0; nt < 4; ++nt) {
            float v0 = a0c[nt][0];
            float v1 = a1c[nt][0];
#pragma unroll
            for (int r = 1; r < 8; ++r) {
                v0 = fmaxf(v0, a0c[nt][r]);
                v1 = fmaxf(v1, a1c[nt][r]);
            }
            v0 = fmaxf(v0, __shfl_xor(v0, 16, 32));
            v1 = fmaxf(v1, __shfl_xor(v1, 16, 32));
            v0 = fmaxf(v0 + bs[nt], 0.0f);
            v1 = fmaxf(v1 + bs[nt], 0.0f);
            if (lane < 16) {
                outp[nt * 16 + nl]       = v0;
                outp[CCH + nt * 16 + nl] = v1;
            }
        }
    }
}

// ---------------------------------------------------------------------------
extern "C" void kernel_launch(void* const* d_in, const int* in_sizes, int n_in,
                              void* d_out, int out_size, void* d_ws, size_t ws_size,
                              hipStream_t stream) {
    const float* coords = (const float*)d_in[0];   // (2, 8192, 3)
    const float* feat   = (const float*)d_in[1];   // (2, 8192, 64)
    const float* Wm     = (const float*)d_in[2];   // (131, 64)
    const float* bias   = (const float*)d_in[3];   // (64,)
    float* out = (float*)d_out;                    // (2, 8192, 64)
    int* idx = (int*)d_ws;                         // 2*8192*16 ints = 1 MB scratch

    knn_kernel<<<64, 256, 0, stream>>>(coords, idx);
    agg_kernel<<<64, 256, 0, stream>>>(coords, feat, Wm, bias, idx, out);
}